// GaussianLayer_16449724744069
// MI455X (gfx1250) — compile-verified
//
#include <hip/hip_runtime.h>

// CDNA5 / gfx1250: wave32, WMMA f32 16x16x4.
typedef float v2f __attribute__((ext_vector_type(2)));
typedef float v8f __attribute__((ext_vector_type(8)));

#define NBINS 199
#define ROWS_PER_WAVE 16
#define WAVES_PER_BLOCK 8
#define BLOCK_THREADS (WAVES_PER_BLOCK * 32)

// D = pdf_tile(16x16) x U(16x16, upper-tri ones) + carry, realized as 4 chained
// V_WMMA_F32_16X16X4_F32 ops per 16-column tile. Carry rides in the C operand.
// Output (417 MB) streams through NT stores: it exceeds the 192 MB L2 and is
// written exactly once, so bypassing the temporal path avoids L2 thrash.
__global__ __launch_bounds__(BLOCK_THREADS)
void gauss_cdf_wmma_kernel(const float2* __restrict__ x,
                           float* __restrict__ out,
                           int nrows) {
  const int lane = threadIdx.x & 31;
  const int waveInBlock = threadIdx.x >> 5;
  const int waveId = blockIdx.x * WAVES_PER_BLOCK + waveInBlock;
  const int rb = waveId * ROWS_PER_WAVE;          // first row of this wave's block
  if (rb >= nrows) return;                        // uniform per wave

  const int n  = lane & 15;                       // column-in-tile (B/C/D lane -> N)
  const int hi = lane >> 4;                       // half-wave select (K/M split)
  const int rowA = rb + n;                        // row whose pdf this lane computes

  // Per-row parameters: mu = x0, sigma = exp(0.5*x1)
  const float2 xv = x[rowA];
  const float mu = xv.x;
  const float lv = xv.y;
  const float LOG2E = 1.4426950408889634f;
  // inv_norm = exp(-0.5*lv) / sqrt(2*pi)
  const float inv_norm =
      __builtin_amdgcn_exp2f(-0.5f * lv * LOG2E) * 0.3989422804014327f;
  // pdf(t) = inv_norm * exp(t^2 * (-0.5 * exp(-lv)))  -> exp2 form
  const float a2 = -0.5f * __builtin_amdgcn_exp2f(-lv * LOG2E) * LOG2E;

  // B operands: upper-triangular ones, U[k][n] = (k <= n). Shared by all tiles.
  // B layout (4x16 f32): lane = N; VGPR0 = K {0|2}, VGPR1 = K {1|3} per half-wave.
  v2f Bk[4];
#pragma unroll
  for (int kk = 0; kk < 4; ++kk) {
    const int k0 = 4 * kk + 2 * hi;
    Bk[kk].x = (k0 <= n) ? 1.0f : 0.0f;
    Bk[kk].y = (k0 + 1 <= n) ? 1.0f : 0.0f;
  }

  // Carry in C/D layout: element r -> rows rb+r (lanes 0-15) / rb+r+8 (lanes 16-31)
  v8f carry = {};

  for (int cb = 0; cb < NBINS; cb += 16) {
    v8f acc = carry;
#pragma unroll
    for (int kk = 0; kk < 4; ++kk) {
      // A layout (16x4 f32): lane = M; VGPR0 = K {0|2}, VGPR1 = K {1|3}.
      const int c0 = cb + 4 * kk + 2 * hi;        // global bin index of A.x
      const float t0 = (float)(c0 - 99) - mu;
      const float t1 = (float)(c0 - 98) - mu;
      v2f A;
      A.x = (c0 < NBINS)     ? inv_norm * __builtin_amdgcn_exp2f(a2 * t0 * t0) : 0.0f;
      A.y = (c0 + 1 < NBINS) ? inv_norm * __builtin_amdgcn_exp2f(a2 * t1 * t1) : 0.0f;
      acc = __builtin_amdgcn_wmma_f32_16x16x4_f32(
          /*neg_a=*/false, A, /*neg_b=*/false, Bk[kk],
          /*c_mod=*/(short)0, acc, /*reuse_a=*/false, /*reuse_b=*/false);
    }

    // Coalesced streaming stores: each half-wave writes 16 consecutive floats
    // per row (64 B segments); NT hint keeps the 417 MB stream out of L2.
    const int col = cb + n;
    const bool ok = col < NBINS;
#pragma unroll
    for (int r = 0; r < 8; ++r) {
      const int row = rb + r + hi * 8;
      if (ok) {
        __builtin_nontemporal_store(acc[r], out + (size_t)row * NBINS + col);
      }
    }

    // New carry = column 15 of D (full prefix incl. old carry), broadcast
    // within each 16-lane half (lane 15 -> 0..15, lane 31 -> 16..31).
#pragma unroll
    for (int r = 0; r < 8; ++r) {
      carry[r] = __shfl(acc[r], 15, 16);
    }
  }
}

extern "C" void kernel_launch(void* const* d_in, const int* in_sizes, int n_in,
                              void* d_out, int out_size, void* d_ws, size_t ws_size,
                              hipStream_t stream) {
  (void)n_in; (void)d_ws; (void)ws_size; (void)out_size;
  const float2* x = (const float2*)d_in[0];
  float* out = (float*)d_out;
  const int nrows = in_sizes[0] / 2;              // x is (B, 2) f32

  const int rowsPerBlock = WAVES_PER_BLOCK * ROWS_PER_WAVE;  // 128
  const int grid = (nrows + rowsPerBlock - 1) / rowsPerBlock;
  gauss_cdf_wmma_kernel<<<grid, BLOCK_THREADS, 0, stream>>>(x, out, nrows);
}